// CarafeUpsampler_26989574488717
// MI455X (gfx1250) — compile-verified
//
#include <hip/hip_runtime.h>
#include <hip/hip_bf16.h>

typedef float v2f __attribute__((ext_vector_type(2)));
typedef float v8f __attribute__((ext_vector_type(8)));

// ---------------------------------------------------------------------------
// Compressor: 1x1 conv C=256 -> 64 as GEMM via V_WMMA_F32_16X16X4_F32.
// out[b, m, n] = sum_k wc[m,k] * x[b,k,n] + bc[m],  n = pixel index (H*W).
// One wave per 16x16 (M x N) tile, K looped in steps of 4.
// ---------------------------------------------------------------------------
__global__ __launch_bounds__(32)
void carafe_compress_wmma(const float* __restrict__ x,    // [B,256,HW]
                          const float* __restrict__ wc,   // [64,256]
                          const float* __restrict__ bc,   // [64]
                          float* __restrict__ comp,       // [B,64,HW]
                          int HW)
{
    const int C = 256;
    const int ntiles = HW >> 4;
    int bid  = blockIdx.x;
    int nt   = bid % ntiles;
    int mt   = (bid / ntiles) & 3;          // 64/16 = 4 M-tiles
    int b    = bid / (ntiles * 4);
    int lane = threadIdx.x;                 // 0..31 (wave32)
    int half = lane >> 4;                   // 0: K lanes {0,1}, 1: K lanes {2,3}
    int l16  = lane & 15;

    int aM   = mt * 16 + l16;               // A row (output channel)
    int nIdx = nt * 16 + l16;               // B column (pixel)
    const float* xb = x + (size_t)b * C * HW;

    v8f acc = {};
    for (int k = 0; k < C; k += 4) {
        int ka = k + half * 2;
        v2f a, bf;
        a[0]  = wc[aM * C + ka];
        a[1]  = wc[aM * C + ka + 1];
        bf[0] = xb[(size_t)ka       * HW + nIdx];
        bf[1] = xb[(size_t)(ka + 1) * HW + nIdx];
        acc = __builtin_amdgcn_wmma_f32_16x16x4_f32(
            false, a, false, bf, (short)0, acc, false, false);
    }

    float* cb = comp + (size_t)b * 64 * HW;
#pragma unroll
    for (int r = 0; r < 8; ++r) {
        int M = mt * 16 + r + half * 8;     // C/D layout: lanes 16-31 hold M+8
        cb[(size_t)M * HW + nIdx] = acc[r] + bc[M];
    }
}

// ---------------------------------------------------------------------------
// Encoder: 3x3 conv 64 -> 36 (pad 1) as im2col GEMM via WMMA.
// M = 36 padded to 48 (3 tiles, zero rows / masked stores), K = 64*9 = 576.
// ---------------------------------------------------------------------------
__global__ __launch_bounds__(32)
void carafe_encode_wmma(const float* __restrict__ comp,  // [B,64,H,W]
                        const float* __restrict__ we,    // [36,64,3,3] = [36][576]
                        const float* __restrict__ be,    // [36]
                        float* __restrict__ mout,        // [B,36,HW]
                        int H, int W)
{
    const int KK = 576;
    const int HW = H * W;
    const int ntiles = HW >> 4;
    int bid  = blockIdx.x;
    int nt   = bid % ntiles;
    int mt   = (bid / ntiles) % 3;          // 3 M-tiles covering 36 rows
    int b    = bid / (ntiles * 3);
    int lane = threadIdx.x;
    int half = lane >> 4;
    int l16  = lane & 15;

    int aM = mt * 16 + l16;
    bool aValid = (aM < 36);
    int n  = nt * 16 + l16;
    int h  = n / W, w = n % W;
    const float* cbase = comp + (size_t)b * 64 * HW;

    v8f acc = {};
    for (int k = 0; k < KK; k += 4) {
        int kk = k + half * 2;
        v2f a, bf;
        a[0] = aValid ? we[aM * KK + kk]     : 0.0f;
        a[1] = aValid ? we[aM * KK + kk + 1] : 0.0f;
#pragma unroll
        for (int t = 0; t < 2; ++t) {
            int kt = kk + t;
            int ic = kt / 9;
            int r  = kt - ic * 9;
            int ky = r / 3, kx = r - ky * 3;
            int hh = h + ky - 1, ww = w + kx - 1;
            float v = 0.0f;
            if ((unsigned)hh < (unsigned)H && (unsigned)ww < (unsigned)W)
                v = cbase[(size_t)ic * HW + hh * W + ww];
            bf[t] = v;
        }
        acc = __builtin_amdgcn_wmma_f32_16x16x4_f32(
            false, a, false, bf, (short)0, acc, false, false);
    }

    float* mb = mout + (size_t)b * 36 * HW;
#pragma unroll
    for (int r = 0; r < 8; ++r) {
        int M = mt * 16 + r + half * 8;
        if (M < 36)
            mb[(size_t)M * HW + n] = acc[r] + be[M];
    }
}

// ---------------------------------------------------------------------------
// Fused pixel-shuffle + softmax + content-aware reassembly (coalesced).
// One block per (b, h, channel-group). Thread layout: w = tid & (W-1),
// csub = tid >> log2(W)  ->  adjacent lanes = adjacent pixels.
//  * 4*W sub-pixel softmaxes computed cooperatively into LDS (mask loads
//    coalesced across w lanes).
//  * Each thread gathers its 3x3 neighborhood (contiguous row segments) and
//    writes the two horizontal outputs as one float2 -> fully coalesced.
// ---------------------------------------------------------------------------
__global__ __launch_bounds__(256)
void carafe_reassemble(const float* __restrict__ x,   // [B,256,H,W]
                       const float* __restrict__ m,   // [B,36,H,W]
                       float* __restrict__ out,       // [B,256,2H,2W]
                       int H, int W, int logW)
{
    __shared__ float wts[128 * 36];        // wts[w][pq][k], max W=128 -> 18 KB
    const int HW  = H * W;
    const int cpb = 256 >> logW;           // channels per block (256/W)

    int bid = blockIdx.x;
    int cb  = bid % W;                     // channel-group (W groups of cpb)
    int h   = (bid / W) % H;
    int b   = bid / (W * H);
    int tid = threadIdx.x;

    // --- cooperative softmax weights for this source row -------------------
    const float* mrow = m + (size_t)b * 36 * HW + (size_t)h * W;
    for (int idx = tid; idx < 4 * W; idx += 256) {
        int ww = idx & (W - 1);            // lane-contiguous in w -> coalesced
        int pq = idx >> logW;              // p*2+q
        float v[9], mx = -3.4e38f;
#pragma unroll
        for (int k = 0; k < 9; ++k) {
            v[k] = mrow[(size_t)(k * 4 + pq) * HW + ww];  // ch = k*S*S + p*S + q
            mx = fmaxf(mx, v[k]);
        }
        float s = 0.0f;
#pragma unroll
        for (int k = 0; k < 9; ++k) { v[k] = __expf(v[k] - mx); s += v[k]; }
        float inv = 1.0f / s;
#pragma unroll
        for (int k = 0; k < 9; ++k) wts[(ww * 4 + pq) * 9 + k] = v[k] * inv;
    }
    __syncthreads();

    // --- reassembly --------------------------------------------------------
    int w    = tid & (W - 1);
    int csub = tid >> logW;
    int c    = cb * cpb + csub;

    const float* xb = x + ((size_t)(b * 256 + c)) * HW;
    float nb[9];
#pragma unroll
    for (int ky = 0; ky < 3; ++ky) {
        int hh = h + ky - 1;
        bool hok = (unsigned)hh < (unsigned)H;
        const float* xrow = xb + (size_t)hh * W;
#pragma unroll
        for (int kx = 0; kx < 3; ++kx) {
            int ww = w + kx - 1;
            float v = 0.0f;
            if (hok && (unsigned)ww < (unsigned)W) v = xrow[ww];
            nb[ky * 3 + kx] = v;
        }
    }

    const float* wl = &wts[w * 36];
    float* ob = out + ((size_t)(b * 256 + c)) * (size_t)(4 * HW);
    int W2 = 2 * W;
#pragma unroll
    for (int p = 0; p < 2; ++p) {
        float2 o;
        float s0 = 0.0f, s1 = 0.0f;
#pragma unroll
        for (int k = 0; k < 9; ++k) {
            s0 += wl[(p * 2 + 0) * 9 + k] * nb[k];
            s1 += wl[(p * 2 + 1) * 9 + k] * nb[k];
        }
        o.x = s0; o.y = s1;
        *(float2*)&ob[(size_t)(2 * h + p) * W2 + 2 * w] = o;   // 8B coalesced
    }
}

// ---------------------------------------------------------------------------
// Host-side launch: 4 stages, ping-pong workspace, final stage -> d_out.
// ws layout (bytes):
//   xA   @ 0         : 2*256*128*128*4 = 33,554,432   (largest intermediate)
//   xB   @ 33554432  : 2*256*64*64*4   =  8,388,608
//   comp @ 41943040  : 2*64*128*128*4  =  8,388,608
//   m    @ 50331648  : 2*36*128*128*4  =  4,718,592   (total ~55 MB)
// ---------------------------------------------------------------------------
extern "C" void kernel_launch(void* const* d_in, const int* in_sizes, int n_in,
                              void* d_out, int out_size, void* d_ws, size_t ws_size,
                              hipStream_t stream) {
    (void)in_sizes; (void)n_in; (void)out_size; (void)ws_size;
    const float* src = (const float*)d_in[0];
    // d_in[1] = guidance: unused by the reference.
    const float* wc[4]; const float* bc[4]; const float* we[4]; const float* be[4];
    for (int i = 0; i < 4; ++i) {
        wc[i] = (const float*)d_in[2 + 4 * i];
        bc[i] = (const float*)d_in[3 + 4 * i];
        we[i] = (const float*)d_in[4 + 4 * i];
        be[i] = (const float*)d_in[5 + 4 * i];
    }

    char* ws = (char*)d_ws;
    float* xA   = (float*)(ws);
    float* xB   = (float*)(ws + 33554432);
    float* comp = (float*)(ws + 41943040);
    float* mbuf = (float*)(ws + 50331648);
    float* outF = (float*)d_out;

    const float* sin[4]  = { src, xA, xB, xA };
    float*       sout[4] = { xA, xB, xA, outF };
    const int    sH[4]   = { 16, 32, 64, 128 };
    const int    sLogW[4] = { 4, 5, 6, 7 };

    const int B = 2;
    for (int s = 0; s < 4; ++s) {
        int H = sH[s], W = sH[s], HW = H * W;
        int ntiles = HW / 16;
        carafe_compress_wmma<<<B * 4 * ntiles, 32, 0, stream>>>(
            sin[s], wc[s], bc[s], comp, HW);
        carafe_encode_wmma<<<B * 3 * ntiles, 32, 0, stream>>>(
            comp, we[s], be[s], mbuf, H, W);
        carafe_reassemble<<<B * H * W, 256, 0, stream>>>(
            sin[s], mbuf, sout[s], H, W, sLogW[s]);
    }
}